// NodeBernNet_15118284881958
// MI455X (gfx1250) — compile-verified
//
#include <hip/hip_runtime.h>
#include <hip/hip_bf16.h>

// ---------------- problem constants ----------------
#define NNODES 100000   // divisible by 16 -> 6250 row tiles
#define NEDGES 1600000
#define CIN    128
#define CC     64       // C_OUT
#define C2     32       // C_OUT/2
#define NCLS   2

typedef float v2f __attribute__((ext_vector_type(2)));
typedef float v8f __attribute__((ext_vector_type(8)));

// ---------------- WMMA helpers (f32 16x16x4, CDNA5) ----------------
// D = A(16x4) * B(4x16) + C(16x16), all f32, wave32.
// A: 2 VGPRs/lane. lanes 0-15: rows 0-15, K = k0,k0+1 ; lanes 16-31: K = k0+2,k0+3
// B: 2 VGPRs/lane. VGPR0: lanes0-15 K=k0, lanes16-31 K=k0+1 ; VGPR1: K=k0+2 / k0+3
// C/D: VGPR i: lanes 0-15 -> row i, lanes 16-31 -> row i+8 ; col = lane&15
__device__ __forceinline__ v8f wmma_f32x4(v2f a, v2f b, v8f c) {
    return __builtin_amdgcn_wmma_f32_16x16x4_f32(
        /*neg_a=*/false, a, /*neg_b=*/false, b,
        /*c_mod=*/(short)0, c, /*reuse_a=*/false, /*reuse_b=*/false);
}

__device__ __forceinline__ v2f frag_a(const float* __restrict__ A, int lda, int k0) {
    int lane = threadIdx.x & 31;
    int row  = lane & 15;
    int kk   = k0 + ((lane >> 4) << 1);
    v2f a;
    a.x = A[row * lda + kk];
    a.y = A[row * lda + kk + 1];
    return a;
}

__device__ __forceinline__ v2f frag_b(const float* __restrict__ B, int ldb, int k0, int cb) {
    int lane = threadIdx.x & 31;
    int col  = cb + (lane & 15);
    int kk   = k0 + (lane >> 4);
    v2f b;
    b.x = B[kk * ldb + col];
    b.y = B[(kk + 2) * ldb + col];
    return b;
}

// ---------------- async global->LDS staging (gfx1250 ASYNCcnt path) ----------------
// LDS aperture: low 32 bits of a generic pointer to __shared__ == LDS byte offset.
__device__ __forceinline__ unsigned lds_off(const void* p) {
    return (unsigned)(uintptr_t)p;
}
// GVS addressing: mem = SADDR(64b sgpr) + VADDR(32b byte off). VDST = LDS byte offset.
__device__ __forceinline__ void async_copy_b128(unsigned lds_byte, unsigned voff,
                                                const void* sbase) {
    asm volatile("global_load_async_to_lds_b128 %0, %1, %2"
                 :: "v"(lds_byte), "v"(voff), "s"(sbase)
                 : "memory");
}
__device__ __forceinline__ void wait_async0() {
    asm volatile("s_wait_asynccnt 0x0" ::: "memory");
}

// ---------------- kernel 1: h = relu(x @ W_in + b_in) ----------------
// grid = N/16 blocks of 128 threads (4 waves). Wave w -> output cols [16w,16w+16).
__global__ void k_gemm_in(const float* __restrict__ x, const float* __restrict__ Win,
                          const float* __restrict__ bin, float* __restrict__ h) {
    __shared__ float sx[16][CIN + 4];   // padded: row stride 528B (16B aligned)
    const int tile = blockIdx.x;
    const int tid  = threadIdx.x;
    const int wave = tid >> 5;

    // async-stage cols 0-31 of the 16x128 x-tile: one b128 per thread (128 * 16B = 2KB)
    {
        int r = tid >> 3, c4 = tid & 7;
        unsigned voff = (unsigned)(r * CIN + c4 * 4) * 4u;
        unsigned ldso = lds_off(&sx[0][0]) + (unsigned)(r * (CIN + 4) + c4 * 4) * 4u;
        async_copy_b128(ldso, voff, x + (size_t)tile * 16 * CIN);
        wait_async0();
    }
    __syncthreads();

    const int cb = wave * 16;
    v8f acc = {};
#pragma unroll
    for (int k0 = 0; k0 < CIN; k0 += 4) {
        v2f a = frag_a(&sx[0][0], CIN + 4, k0);
        v2f b = frag_b(Win, CC, k0, cb);
        acc = wmma_f32x4(a, b, acc);
    }
    const int lane = tid & 31;
    const int col  = cb + (lane & 15);
    const float bias = bin[col];
    const int rbase = tile * 16 + ((lane >> 4) << 3);
#pragma unroll
    for (int i = 0; i < 8; ++i) {
        float v = acc[i] + bias;
        h[(size_t)(rbase + i) * CC + col] = fmaxf(v, 0.f);
    }
}

// ---------------- kernel 2: zero fill ----------------
__global__ void k_zero(float* __restrict__ p, int n) {
    int i = blockIdx.x * blockDim.x + threadIdx.x;
    if (i < n) p[i] = 0.f;
}

// ---------------- kernel 3: SpMM  out += vals[e] * hin[src[e]] scattered to dst[e] ----
// 16 lanes per edge (2 edges/wave); b128 gather of L2-resident rows, f32 atomics scatter.
__global__ void k_spmm(const int* __restrict__ src, const int* __restrict__ dst,
                       const float* __restrict__ vals,
                       const float* __restrict__ hin, float* __restrict__ hout) {
    int gtid = blockIdx.x * blockDim.x + threadIdx.x;
    int e    = gtid >> 4;
    if (e >= NEDGES) return;
    int l = gtid & 15;
    int s = src[e];
    int d = dst[e];
    float v = vals[e];
    float4 hv = ((const float4*)(hin + (size_t)s * CC))[l];
    float* o = hout + (size_t)d * CC + l * 4;
    atomicAdd(o + 0, v * hv.x);
    atomicAdd(o + 1, v * hv.y);
    atomicAdd(o + 2, v * hv.z);
    atomicAdd(o + 3, v * hv.w);
}

// ---------------- kernel 4: Bernstein/theta combine -> poly0, poly1 ----------------
__constant__ float cBC[4][4] = {
    {1.f, -3.f,  3.f, -1.f},
    {0.f,  3.f, -6.f,  3.f},
    {0.f,  0.f,  3.f, -3.f},
    {0.f,  0.f,  0.f,  1.f},
};

__global__ void k_poly(const float* __restrict__ t0, const float* __restrict__ t1,
                       const float* __restrict__ t2, const float* __restrict__ t3,
                       const float* __restrict__ thetas,
                       float* __restrict__ p0, float* __restrict__ p1) {
    int i = blockIdx.x * blockDim.x + threadIdx.x;
    if (i >= NNODES * CC) return;
    // coef[f][k] = sum_j thetas[f,j] * BCOEF[j,k]
    float cf0[4], cf1[4];
#pragma unroll
    for (int k = 0; k < 4; ++k) {
        cf0[k] = thetas[0] * cBC[0][k] + thetas[1] * cBC[1][k] +
                 thetas[2] * cBC[2][k] + thetas[3] * cBC[3][k];
        cf1[k] = thetas[4] * cBC[0][k] + thetas[5] * cBC[1][k] +
                 thetas[6] * cBC[2][k] + thetas[7] * cBC[3][k];
    }
    float v0 = t0[i], v1 = t1[i], v2 = t2[i], v3 = t3[i];
    p0[i] = cf0[0] * v0 + cf0[1] * v1 + cf0[2] * v2 + cf0[3] * v3;
    p1[i] = cf1[0] * v0 + cf1[1] * v1 + cf1[2] * v2 + cf1[3] * v3;
}

// ---------------- kernel 5: fused attention + classifier head ----------------
// per 16-node tile: poly_proj (2 GEMMs sharing B), x_proj GEMM, tanh, vc-dot,
// softmax over F=2, weighted sum, then relu(W1), relu(W2), W3 -> logits.
__global__ void k_attn(const float* __restrict__ h,
                       const float* __restrict__ p0g, const float* __restrict__ p1g,
                       const float* __restrict__ Wb, const float* __restrict__ bWb,
                       const float* __restrict__ Wx, const float* __restrict__ bWx,
                       const float* __restrict__ vc,
                       const float* __restrict__ W1, const float* __restrict__ b1,
                       const float* __restrict__ W2, const float* __restrict__ b2,
                       const float* __restrict__ W3, const float* __restrict__ b3,
                       float* __restrict__ out) {
    __shared__ float sP0[16][CC + 4];   // row stride 272B (16B aligned)
    __shared__ float sP1[16][CC + 4];
    __shared__ float sH [16][CC + 4];   // h tile; reused for res
    __shared__ float sT0[16][CC + 4];   // tanh(f=0); reused for y1
    __shared__ float sT1[16][CC + 4];
    __shared__ float sY2[16][C2 + 4];
    __shared__ float sAttn[16][2];

    const int tile = blockIdx.x;
    const int tid  = threadIdx.x;
    const int wave = tid >> 5;
    const int lane = tid & 31;
    const size_t base = (size_t)tile * 16 * CC;

    // async-stage cols 0-15 of the three 16x64 tiles (poly0, poly1, h), b128 per lane.
    // Each wave covers 8 rows of one tile per issue; saddr is wave-uniform.
    {
        int rem = ((wave & 1) << 5) | lane;      // 0..63 within a tile
        int r = rem >> 2, c4 = rem & 3;
        unsigned voff = (unsigned)(r * CC + c4 * 4) * 4u;
        unsigned loff = (unsigned)(r * (CC + 4) + c4 * 4) * 4u;
        const float* g0 = (wave < 2) ? p0g : p1g;
        unsigned     l0 = ((wave < 2) ? lds_off(&sP0[0][0]) : lds_off(&sP1[0][0])) + loff;
        async_copy_b128(l0, voff, g0 + base);
        if (wave < 2)
            async_copy_b128(lds_off(&sH[0][0]) + loff, voff, h + base);
        wait_async0();
    }
    __syncthreads();

    const int cb = wave * 16;
    v8f cp0 = {}, cp1 = {}, cxp = {};
#pragma unroll
    for (int k0 = 0; k0 < CC; k0 += 4) {
        v2f a0 = frag_a(&sP0[0][0], CC + 4, k0);
        v2f a1 = frag_a(&sP1[0][0], CC + 4, k0);
        v2f ah = frag_a(&sH[0][0],  CC + 4, k0);
        v2f bb = frag_b(Wb, CC, k0, cb);   // shared across both filters
        v2f bx = frag_b(Wx, CC, k0, cb);
        cp0 = wmma_f32x4(a0, bb, cp0);
        cp1 = wmma_f32x4(a1, bb, cp1);
        cxp = wmma_f32x4(ah, bx, cxp);
    }
    {
        const int col = cb + (lane & 15);
        const float bsum = bWb[col] + bWx[col];
        const int rb = (lane >> 4) << 3;
#pragma unroll
        for (int i = 0; i < 8; ++i) {
            sT0[rb + i][col] = tanhf(cp0[i] + cxp[i] + bsum);
            sT1[rb + i][col] = tanhf(cp1[i] + cxp[i] + bsum);
        }
    }
    __syncthreads();

    // attention scores + softmax over F=2 (one thread per row)
    if (tid < 16) {
        float s0 = 0.f, s1 = 0.f;
#pragma unroll
        for (int c = 0; c < CC; ++c) {
            float v = vc[c];
            s0 += sT0[tid][c] * v;
            s1 += sT1[tid][c] * v;
        }
        float m = fmaxf(s0, s1);
        float e0 = __expf(s0 - m), e1 = __expf(s1 - m);
        float inv = 1.f / (e0 + e1);
        sAttn[tid][0] = e0 * inv;
        sAttn[tid][1] = e1 * inv;
    }
    __syncthreads();

    // res = a0*poly0 + a1*poly1  (overwrite sH; h no longer needed)
    for (int i = tid; i < 16 * CC; i += 128) {
        int r = i >> 6, c = i & 63;
        sH[r][c] = sAttn[r][0] * sP0[r][c] + sAttn[r][1] * sP1[r][c];
    }
    __syncthreads();

    // y1 = relu(res @ W1 + b1)  -> sT0
    {
        v8f acc = {};
#pragma unroll
        for (int k0 = 0; k0 < CC; k0 += 4) {
            v2f a = frag_a(&sH[0][0], CC + 4, k0);
            v2f b = frag_b(W1, CC, k0, cb);
            acc = wmma_f32x4(a, b, acc);
        }
        const int col = cb + (lane & 15);
        const float bias = b1[col];
        const int rb = (lane >> 4) << 3;
#pragma unroll
        for (int i = 0; i < 8; ++i)
            sT0[rb + i][col] = fmaxf(acc[i] + bias, 0.f);
    }
    __syncthreads();

    // y2 = relu(y1 @ W2 + b2)  [16,64]x[64,32] -> waves 0,1 only
    if (wave < 2) {
        v8f acc = {};
#pragma unroll
        for (int k0 = 0; k0 < CC; k0 += 4) {
            v2f a = frag_a(&sT0[0][0], CC + 4, k0);
            v2f b = frag_b(W2, C2, k0, wave * 16);
            acc = wmma_f32x4(a, b, acc);
        }
        const int col = wave * 16 + (lane & 15);
        const float bias = b2[col];
        const int rb = (lane >> 4) << 3;
#pragma unroll
        for (int i = 0; i < 8; ++i)
            sY2[rb + i][col] = fmaxf(acc[i] + bias, 0.f);
    }
    __syncthreads();

    // y3 = y2 @ W3 + b3  [16,32]x[32,2] -> 32 threads
    if (tid < 32) {
        int r = tid >> 1, cls = tid & 1;
        float s = b3[cls];
#pragma unroll
        for (int k = 0; k < C2; ++k)
            s += sY2[r][k] * W3[k * NCLS + cls];
        out[((size_t)tile * 16 + r) * NCLS + cls] = s;
    }
}

// ---------------- host launch ----------------
extern "C" void kernel_launch(void* const* d_in, const int* in_sizes, int n_in,
                              void* d_out, int out_size, void* d_ws, size_t ws_size,
                              hipStream_t stream) {
    const float* x      = (const float*)d_in[0];
    const int*   esrc   = (const int*)  d_in[1];
    const int*   edst   = (const int*)  d_in[2];
    const float* evals  = (const float*)d_in[3];
    const float* Win    = (const float*)d_in[4];
    const float* bin    = (const float*)d_in[5];
    const float* thetas = (const float*)d_in[6];
    const float* Wb     = (const float*)d_in[7];
    const float* bWb    = (const float*)d_in[8];
    const float* Wx     = (const float*)d_in[9];
    const float* bWx    = (const float*)d_in[10];
    const float* vc     = (const float*)d_in[11];
    const float* W1     = (const float*)d_in[12];
    const float* b1     = (const float*)d_in[13];
    const float* W2     = (const float*)d_in[14];
    const float* b2     = (const float*)d_in[15];
    const float* W3     = (const float*)d_in[16];
    const float* b3     = (const float*)d_in[17];
    float* out = (float*)d_out;

    // workspace layout: 6 x [N,64] f32 buffers = 153.6 MB
    float* h  = (float*)d_ws;
    float* t1 = h  + (size_t)NNODES * CC;
    float* t2 = t1 + (size_t)NNODES * CC;
    float* t3 = t2 + (size_t)NNODES * CC;
    float* p0 = t3 + (size_t)NNODES * CC;
    float* p1 = p0 + (size_t)NNODES * CC;

    // 1) input transform (WMMA f32)
    k_gemm_in<<<NNODES / 16, 128, 0, stream>>>(x, Win, bin, h);

    // 2) zero Tx1..Tx3 accumulators, then 3 chained SpMMs
    {
        int n = 3 * NNODES * CC;
        k_zero<<<(n + 255) / 256, 256, 0, stream>>>(t1, n);
    }
    const int spmm_grid = (NEDGES * 16 + 255) / 256;   // 16 lanes per edge
    k_spmm<<<spmm_grid, 256, 0, stream>>>(esrc, edst, evals, h,  t1);
    k_spmm<<<spmm_grid, 256, 0, stream>>>(esrc, edst, evals, t1, t2);
    k_spmm<<<spmm_grid, 256, 0, stream>>>(esrc, edst, evals, t2, t3);

    // 3) Bernstein basis + theta combine
    {
        int n = NNODES * CC;
        k_poly<<<(n + 255) / 256, 256, 0, stream>>>(h, t1, t2, t3, thetas, p0, p1);
    }

    // 4) fused attention + classifier head -> logits
    k_attn<<<NNODES / 16, 128, 0, stream>>>(h, p0, p1, Wb, bWb, Wx, bWx, vc,
                                            W1, b1, W2, b2, W3, b3, out);
}